// AttentionSpeller_42125039239193
// MI455X (gfx1250) — compile-verified
//
#include <hip/hip_runtime.h>

// ---------------------------------------------------------------------------
// AttentionSpeller for MI455X (gfx1250, wave32).
// fp32 everywhere (matches reference). Matrix work on V_WMMA_F32_16X16X4_F32.
// B operand is pre-packed in k-pair layout so each lane's B fragment is one
// coalesced 8-byte load:  Bp[(k>>1)*2N + 2n + (k&1)] = W[n][k].
// ---------------------------------------------------------------------------

typedef float v2f __attribute__((ext_vector_type(2)));
typedef float v8f __attribute__((ext_vector_type(8)));

#define WAVES_PER_BLOCK 8   // 256 threads, 8 waves; each wave owns a 16x64 strip

// Branch-free WMMA k-loop for NT tiles (NT is compile-time; nt is wave-uniform).
template<int NT>
__device__ __forceinline__ void wmma_kloop(v8f* acc, const float* __restrict__ Arow,
                                           const float* __restrict__ Bbase,
                                           int K, size_t twoN, int half)
{
    #pragma unroll 4
    for (int k = 0; k < K; k += 4) {
        v2f a = *(const v2f*)(Arow + k);                       // A: {A[m][k+2h], A[m][k+2h+1]}
        const float* brow = Bbase + (size_t)((k >> 1) + half) * twoN;
        #pragma unroll
        for (int t = 0; t < NT; ++t) {
            v2f b = *(const v2f*)(brow + (t << 5));            // one b64, coalesced
            acc[t] = __builtin_amdgcn_wmma_f32_16x16x4_f32(
                false, a, false, b, (short)0, acc[t], false, false);
        }
    }
}

template<int NT>
__device__ __forceinline__ void run_passes(v8f* acc,
    const float* __restrict__ A1, int lda1, const float* __restrict__ B1, int K1,
    const float* __restrict__ A2, int lda2, const float* __restrict__ B2, int K2,
    int m0, int l16, int half, int n0, size_t twoN)
{
    const float* Arow1 = A1 + (size_t)(m0 + l16) * lda1 + 2 * half;
    const float* Bb1   = B1 + 2 * (size_t)(n0 + l16);
    wmma_kloop<NT>(acc, Arow1, Bb1, K1, twoN, half);
    if (A2) {
        const float* Arow2 = A2 + (size_t)(m0 + l16) * lda2 + 2 * half;
        const float* Bb2   = B2 + 2 * (size_t)(n0 + l16);
        wmma_kloop<NT>(acc, Arow2, Bb2, K2, twoN, half);
    }
}

// Generic f32 WMMA GEMM:  D[M x N] = A1*B1 (+ A2*B2) (+ Cadd) (+ biasA + biasB)
// A row-major (lda); B in packed k-pair layout (see header).  M, N multiples of 16.
__global__ __launch_bounds__(256) void gemm_f32_wmma(
    const float* __restrict__ A1, int lda1, const float* __restrict__ B1, int K1,
    const float* __restrict__ A2, int lda2, const float* __restrict__ B2, int K2,
    const float* __restrict__ Cadd, int ldc,
    const float* __restrict__ biasA, const float* __restrict__ biasB,
    float* __restrict__ D, int ldd, int N)
{
    const int lane = threadIdx.x & 31;
    const int wave = threadIdx.x >> 5;
    const int half = lane >> 4;          // half-wave: splits K (A/B) and M (C/D)
    const int l16  = lane & 15;
    const int m0   = blockIdx.y << 4;
    const int n0   = (blockIdx.x * WAVES_PER_BLOCK + wave) << 6;
    if (n0 >= N) return;                 // wave-uniform; kernel has no barriers
    int nt = (N - n0) >> 4; if (nt > 4) nt = 4;   // N % 16 == 0 in all uses

    v8f acc[4];
    #pragma unroll
    for (int t = 0; t < 4; ++t)
        #pragma unroll
        for (int v = 0; v < 8; ++v) acc[t][v] = 0.0f;

    if (Cadd) {
        #pragma unroll
        for (int t = 0; t < 4; ++t) if (t < nt) {
            const float* cp = Cadd + (size_t)(m0 + 8 * half) * ldc + n0 + (t << 4) + l16;
            #pragma unroll
            for (int v = 0; v < 8; ++v) acc[t][v] = cp[(size_t)v * ldc];
        }
    }

    const size_t twoN = 2 * (size_t)N;
    switch (nt) {
    case 4: run_passes<4>(acc, A1, lda1, B1, K1, A2, lda2, B2, K2, m0, l16, half, n0, twoN); break;
    case 3: run_passes<3>(acc, A1, lda1, B1, K1, A2, lda2, B2, K2, m0, l16, half, n0, twoN); break;
    case 2: run_passes<2>(acc, A1, lda1, B1, K1, A2, lda2, B2, K2, m0, l16, half, n0, twoN); break;
    default: run_passes<1>(acc, A1, lda1, B1, K1, A2, lda2, B2, K2, m0, l16, half, n0, twoN); break;
    }

    #pragma unroll
    for (int t = 0; t < 4; ++t) if (t < nt) {
        const int n = n0 + (t << 4) + l16;
        float bv = 0.0f;
        if (biasA) bv += biasA[n];
        if (biasB) bv += biasB[n];
        float* dp = D + (size_t)(m0 + 8 * half) * ldd + n;
        #pragma unroll
        for (int v = 0; v < 8; ++v) dp[(size_t)v * ldd] = acc[t][v] + bv;
    }
}

// Transpose + k-pair pack:  out[(k>>1)*2*Nn + 2*n + (k&1)] = in[n*K + k]
// in: weight (Nn rows x K cols).  One-time per launch.
__global__ __launch_bounds__(256) void transpose_pack2_k(
    const float* __restrict__ in, float* __restrict__ out, int Nn, int K)
{
    __shared__ float tile[32][33];
    const int tx = threadIdx.x & 31;
    const int ty = threadIdx.x >> 5;        // 0..7
    const int k0 = blockIdx.x << 5;
    const int n0 = blockIdx.y << 5;
    #pragma unroll
    for (int i = 0; i < 32; i += 8) {
        int n = n0 + ty + i, k = k0 + tx;
        if (n < Nn && k < K) tile[ty + i][tx] = in[(size_t)n * K + k];
    }
    __syncthreads();
    const size_t twoN = 2 * (size_t)Nn;
    #pragma unroll
    for (int i = 0; i < 32; i += 8) {
        int k = k0 + ty + i, n = n0 + tx;
        if (k < K && n < Nn)
            out[(size_t)(k >> 1) * twoN + 2 * n + (k & 1)] = tile[tx][ty + i];
    }
}

__device__ __forceinline__ float sigmoidf_(float x) { return 1.0f / (1.0f + __expf(-x)); }

// Fused: layer-1 LSTM cell activation + dot-product attention + softmax + context.
// One workgroup (512 threads) per batch element b.  g1 = precomputed gates (B x 2048).
__global__ __launch_bounds__(512) void attn_step_k(
    const float* __restrict__ g1, const float* __restrict__ listener, // (U=1024, B=32, H=512)
    float* __restrict__ h1, float* __restrict__ c1,
    float* __restrict__ ctx_out)                                      // (B, H) slice
{
    __shared__ float sh[512];     // h1_new for this b
    __shared__ float sw[1024];    // scores -> weights
    __shared__ float red[512];
    const int b = blockIdx.x;
    const int tid = threadIdx.x;

    // ---- LSTM cell (gate order i,f,g,o) ----
    {
        const float* gb = g1 + (size_t)b * 2048;
        float iv = sigmoidf_(gb[tid]);
        float fv = sigmoidf_(gb[512 + tid]);
        float gv = tanhf(gb[1024 + tid]);
        float ov = sigmoidf_(gb[1536 + tid]);
        float cn = fv * c1[b * 512 + tid] + iv * gv;
        float hn = ov * tanhf(cn);
        c1[b * 512 + tid] = cn;
        h1[b * 512 + tid] = hn;
        sh[tid] = hn;
    }
    __syncthreads();

    // ---- scores[u] = listener[u,b,:] . h1[b,:]  (2 u per thread) ----
    for (int u = tid; u < 1024; u += 512) {
        const float4* lp = (const float4*)(listener + ((size_t)u * 32 + b) * 512);
        const float4* hp = (const float4*)sh;
        float s = 0.0f;
        #pragma unroll 4
        for (int q = 0; q < 128; ++q) {
            float4 lv = lp[q], hv = hp[q];
            s += lv.x * hv.x + lv.y * hv.y + lv.z * hv.z + lv.w * hv.w;
        }
        sw[u] = s;
    }
    __syncthreads();

    // ---- softmax over u (1024) ----
    red[tid] = fmaxf(sw[tid], sw[tid + 512]);
    __syncthreads();
    for (int s = 256; s > 0; s >>= 1) {
        if (tid < s) red[tid] = fmaxf(red[tid], red[tid + s]);
        __syncthreads();
    }
    const float mx = red[0];
    __syncthreads();
    float e0 = __expf(sw[tid] - mx);
    float e1 = __expf(sw[tid + 512] - mx);
    sw[tid] = e0; sw[tid + 512] = e1;
    red[tid] = e0 + e1;
    __syncthreads();
    for (int s = 256; s > 0; s >>= 1) {
        if (tid < s) red[tid] += red[tid + s];
        __syncthreads();
    }
    const float inv = 1.0f / red[0];
    __syncthreads();
    sw[tid] *= inv; sw[tid + 512] *= inv;
    __syncthreads();

    // ---- ctx[h] = sum_u w[u] * listener[u,b,h]  (coalesced in h per u) ----
    const float* lp = listener + (size_t)b * 512 + tid;
    float acc = 0.0f;
    #pragma unroll 4
    for (int u = 0; u < 1024; ++u)
        acc += sw[u] * lp[(size_t)u * (32 * 512)];
    ctx_out[b * 512 + tid] = acc;
}

// Layer-2 LSTM cell (elementwise over B*H = 16384)
__global__ __launch_bounds__(256) void cell2_k(
    const float* __restrict__ g2, float* __restrict__ c2,
    float* __restrict__ h2, float* __restrict__ hs_t)
{
    const int idx = blockIdx.x * 256 + threadIdx.x;   // 0..16383
    const int b = idx >> 9, j = idx & 511;
    const float* gb = g2 + (size_t)b * 2048;
    float iv = sigmoidf_(gb[j]);
    float fv = sigmoidf_(gb[512 + j]);
    float gv = tanhf(gb[1024 + j]);
    float ov = sigmoidf_(gb[1536 + j]);
    float cn = fv * c2[idx] + iv * gv;
    float hn = ov * tanhf(cn);
    c2[idx] = cn;
    h2[idx] = hn;
    hs_t[idx] = hn;
}

// In-place row softmax over N (= V = 10000); one 256-thread block per row.
__global__ __launch_bounds__(256) void softmax_rows_k(float* __restrict__ x, int N)
{
    float* row = x + (size_t)blockIdx.x * N;
    const int tid = threadIdx.x;
    __shared__ float red[256];
    float mx = -3.402823466e38f;
    for (int i = tid; i < N; i += 256) mx = fmaxf(mx, row[i]);
    red[tid] = mx; __syncthreads();
    for (int s = 128; s > 0; s >>= 1) { if (tid < s) red[tid] = fmaxf(red[tid], red[tid + s]); __syncthreads(); }
    mx = red[0]; __syncthreads();
    float sum = 0.0f;
    for (int i = tid; i < N; i += 256) sum += __expf(row[i] - mx);
    red[tid] = sum; __syncthreads();
    for (int s = 128; s > 0; s >>= 1) { if (tid < s) red[tid] += red[tid + s]; __syncthreads(); }
    const float inv = 1.0f / red[0];
    for (int i = tid; i < N; i += 256) row[i] = __expf(row[i] - mx) * inv;
}

// ---------------------------------------------------------------------------
extern "C" void kernel_launch(void* const* d_in, const int* in_sizes, int n_in,
                              void* d_out, int out_size, void* d_ws, size_t ws_size,
                              hipStream_t stream)
{
    (void)in_sizes; (void)n_in; (void)out_size; (void)ws_size;
    const float* inputs   = (const float*)d_in[0];   // (128,32,512)
    const float* h0       = (const float*)d_in[1];   // (2,32,512)
    const float* c0       = (const float*)d_in[2];
    const float* listener = (const float*)d_in[3];   // (1024,32,512)
    const float* W_ih1    = (const float*)d_in[4];   // (2048,512)
    const float* W_hh1    = (const float*)d_in[5];   // (2048,512)
    const float* b_ih1    = (const float*)d_in[6];
    const float* b_hh1    = (const float*)d_in[7];
    const float* W_ih2    = (const float*)d_in[8];   // (2048,512)
    const float* W_hh2    = (const float*)d_in[9];
    const float* b_ih2    = (const float*)d_in[10];
    const float* b_hh2    = (const float*)d_in[11];
    const float* W_out    = (const float*)d_in[12];  // (10000,512)
    const float* b_out    = (const float*)d_in[13];
    float* out = (float*)d_out;                      // (128,32,10000)

    const int T = 128, B = 32, H = 512, V = 10000;
    const int G = 4 * H;                             // 2048
    float* ws = (float*)d_ws;
    size_t off = 0;
    auto alloc = [&](size_t n) { float* p = ws + off; off += n; return p; };
    float* Wih1t = alloc((size_t)H * G);
    float* Whh1t = alloc((size_t)H * G);
    float* Wih2t = alloc((size_t)H * G);
    float* Whh2t = alloc((size_t)H * G);
    float* Woutt = alloc((size_t)H * V);
    float* X1    = alloc((size_t)T * B * G);
    float* ctxs  = alloc((size_t)T * B * H);
    float* hs    = alloc((size_t)T * B * H);
    float* g1    = alloc((size_t)B * G);
    float* g2    = alloc((size_t)B * G);
    float* h1    = alloc((size_t)B * H);
    float* c1    = alloc((size_t)B * H);
    float* h2    = alloc((size_t)B * H);
    float* c2    = alloc((size_t)B * H);

    // One-time weight transposes into packed k-pair layout.
    transpose_pack2_k<<<dim3(H / 32, G / 32), 256, 0, stream>>>(W_ih1, Wih1t, G, H);
    transpose_pack2_k<<<dim3(H / 32, G / 32), 256, 0, stream>>>(W_hh1, Whh1t, G, H);
    transpose_pack2_k<<<dim3(H / 32, G / 32), 256, 0, stream>>>(W_ih2, Wih2t, G, H);
    transpose_pack2_k<<<dim3(H / 32, G / 32), 256, 0, stream>>>(W_hh2, Whh2t, G, H);
    transpose_pack2_k<<<dim3(H / 32, (V + 31) / 32), 256, 0, stream>>>(W_out, Woutt, V, H);

    // Seed recurrent state from h0/c0 (capture-safe async d2d copies).
    hipMemcpyAsync(h1, h0,         (size_t)B * H * 4, hipMemcpyDeviceToDevice, stream);
    hipMemcpyAsync(c1, c0,         (size_t)B * H * 4, hipMemcpyDeviceToDevice, stream);
    hipMemcpyAsync(h2, h0 + B * H, (size_t)B * H * 4, hipMemcpyDeviceToDevice, stream);
    hipMemcpyAsync(c2, c0 + B * H, (size_t)B * H * 4, hipMemcpyDeviceToDevice, stream);

    // X1 = inputs @ W_ih1^T + (b_ih1 + b_hh1)   -- fully parallel over T*B rows
    gemm_f32_wmma<<<dim3((G + 511) / 512, (T * B) / 16), 256, 0, stream>>>(
        inputs, H, Wih1t, H, nullptr, 0, nullptr, 0,
        nullptr, 0, b_ih1, b_hh1, X1, G, G);

    // Scan 1: recurrent GEMM + fused cell/attention per timestep.
    for (int t = 0; t < T; ++t) {
        gemm_f32_wmma<<<dim3((G + 511) / 512, B / 16), 256, 0, stream>>>(
            h1, H, Whh1t, H, nullptr, 0, nullptr, 0,
            X1 + (size_t)t * B * G, G, nullptr, nullptr, g1, G, G);
        attn_step_k<<<B, 512, 0, stream>>>(g1, listener, h1, c1,
                                           ctxs + (size_t)t * B * H);
    }

    // Scan 2: dual-K GEMM (ctx*W_ih2^T + h2*W_hh2^T + biases) + cell per timestep.
    for (int t = 0; t < T; ++t) {
        gemm_f32_wmma<<<dim3((G + 511) / 512, B / 16), 256, 0, stream>>>(
            ctxs + (size_t)t * B * H, H, Wih2t, H, h2, H, Whh2t, H,
            nullptr, 0, b_ih2, b_hh2, g2, G, G);
        cell2_k<<<(B * H) / 256, 256, 0, stream>>>(g2, c2, h2,
                                                   hs + (size_t)t * B * H);
    }

    // Logits directly into d_out, then in-place softmax over V.
    gemm_f32_wmma<<<dim3((V + 511) / 512, (T * B) / 16), 256, 0, stream>>>(
        hs, H, Woutt, H, nullptr, 0, nullptr, 0,
        nullptr, 0, b_out, nullptr, out, V, V);
    softmax_rows_k<<<T * B, 256, 0, stream>>>(out, V);
}